// QuantSoftmax_15719580304039
// MI455X (gfx1250) — compile-verified
//
#include <hip/hip_runtime.h>
#include <stdint.h>

// ---------------------------------------------------------------------------
// Quantized softmax over C=128 channels of an int8-valued [B,C,H,W] tensor.
//
// Roofline (MI455X): traffic = 105MB int32 in + 105MB f32 out ~= 210MB at
// 23.3 TB/s -> ~9us floor; VALU work ~30 ops/elem is noise. Pure data-
// movement kernel: each workgroup TDM-loads a [C=128 x TILE=256] int32 slab
// (channel stride = 6400) into LDS, runs the 3 channel passes (max, exp-LUT
// sum, requantize) column-wise in LDS, requantizes IN PLACE, and TDM-stores
// the f32 slab back out. The piecewise-quantized exp collapses to a 256-entry
// LUT because (qmax - q) takes only 256 integer values.
// ---------------------------------------------------------------------------

#define CCH   128      // channels
#define HWPX  6400     // 80*80 pixels per (b,c) plane
#define TILE  256      // pixels per workgroup
#define BLOCK 256      // one thread per pixel (8 wave32s)

typedef uint32_t u32;
typedef unsigned int uv4 __attribute__((ext_vector_type(4)));
typedef int iv8 __attribute__((ext_vector_type(8)));
typedef int iv4 __attribute__((ext_vector_type(4)));

__device__ __forceinline__ float quant_q(float y, float scale) {
    float q = rintf(y / scale);                    // round-half-even == jnp.round
    return fminf(fmaxf(q, -32768.0f), 32767.0f);   // clip to int16 range
}

// piecewise quantized exp(x) -> int16-valued float, EXP_OUT_SCALE = 2/65535
__device__ float exp_pw_q(float x) {
    const float S = (float)(2.0 / 65535.0);
    if (x > 1.0f)  return quant_q(expf(1.0f), S);
    if (x >= 0.0f) return quant_q(1.0f + x * (expf(1.0f) - 1.0f), S);
    if (x >= -10.0f) {   // 256-entry table region (-10, 0]
        float fi = fminf(fmaxf(rintf((x + 10.0f) * 25.5f), 0.0f), 255.0f);
        return quant_q(expf(-10.0f + fi * (float)(10.0 / 255.0)), S);
    }
    if (x >= -20.0f) {   // 256-entry table region (-20, -10)
        float fi = fminf(fmaxf(rintf((x + 20.0f) * 25.5f), 0.0f), 255.0f);
        return quant_q(expf(-20.0f + fi * (float)(10.0 / 255.0)), S);
    }
    if (x >= -40.0f) {
        float y0 = expf(-40.0f), y1 = expf(-20.0f);
        return quant_q(y0 + (x + 40.0f) * ((y1 - y0) * (1.0f / 20.0f)), S);
    }
    return quant_q(expf(-40.0f), S);
}

// piecewise quantized 1/v -> int16-valued float, RECIP_OUT_SCALE = 20/65535
__device__ float recip_pw_q(float v) {
    const float S = (float)(20.0 / 65535.0);
    if (v > 700.0f)  return quant_q(1.0f / 700.0f, S);
    if (v >= 500.0f)
        return quant_q(1.0f / 500.0f +
                       (v - 500.0f) * (float)(((1.0 / 700.0) - (1.0 / 500.0)) / 200.0), S);
    if (v >= 250.0f) {
        float fi = fminf(fmaxf(rintf((v - 250.0f) * (float)(255.0 / 250.0)), 0.0f), 255.0f);
        return quant_q(1.0f / (250.0f + fi * (float)(250.0 / 255.0)), S);
    }
    if (v >= 0.1f) {     // the region actually hit: v in [~1, 128]
        float fi = fminf(fmaxf(rintf((v - 0.1f) * (float)(255.0 / 249.9)), 0.0f), 255.0f);
        return quant_q(1.0f / (0.1f + fi * (float)(249.9 / 255.0)), S);
    }
    if (v >= 0.001f)
        return quant_q(1000.0f + (v - 0.001f) * (float)((10.0 - 1000.0) / 0.099), S);
    return quant_q(1000.0f, S);
}

#if __has_builtin(__builtin_amdgcn_tensor_load_to_lds)
// Build the shared D# for a [CCH x TILE] 4-byte slab with row stride HWPX.
// ga = global byte address of the tile start; lds = LDS byte offset.
__device__ __forceinline__ void make_desc(uint64_t ga, u32 lds, uv4& g0, iv8& g1) {
    g0.x = 1u;                                               // count=1 valid user descriptor
    g0.y = lds;                                              // lds_addr (bytes)
    g0.z = (u32)ga;                                          // global_addr[31:0]
    g0.w = (u32)((ga >> 32) & 0x01FFFFFFu) | 0x80000000u;    // addr[56:32] | type=2 (image)
    g1[0] = (int)(2u << 16);                                 // data_size = 4 bytes
    g1[1] = (int)((u32)(HWPX & 0xFFFF) << 16);               // tensor_dim0[15:0]
    g1[2] = (int)((u32)(HWPX >> 16) |
                  ((u32)(CCH & 0xFFFF) << 16));              // dim0[31:16] | dim1[15:0]
    g1[3] = (int)((u32)(CCH >> 16) | ((u32)TILE << 16));     // dim1[31:16] | tile_dim0=256
    g1[4] = (int)CCH;                                        // tile_dim1=128 (tile_dim2=0)
    g1[5] = (int)HWPX;                                       // tensor_dim0_stride lo32 (=6400)
    g1[6] = 0;                                               // stride0 hi16 | stride1 lo16
    g1[7] = 0;                                               // stride1 hi32
}

__device__ __forceinline__ void tdm_load(uint64_t ga, u32 lds) {
    uv4 g0; iv8 g1; make_desc(ga, lds, g0, g1);
    iv4 g2 = {0, 0, 0, 0}, g3 = {0, 0, 0, 0};
#if __has_include(<hip/amd_detail/amd_gfx1250_TDM.h>)
    iv8 g4 = {0, 0, 0, 0, 0, 0, 0, 0};
    __builtin_amdgcn_tensor_load_to_lds(g0, g1, g2, g3, g4, 0);   // 6-arg toolchain
#else
    __builtin_amdgcn_tensor_load_to_lds(g0, g1, g2, g3, 0);       // 5-arg ROCm 7.2
#endif
}

__device__ __forceinline__ void tdm_store(uint64_t ga, u32 lds) {
    uv4 g0; iv8 g1; make_desc(ga, lds, g0, g1);
    iv4 g2 = {0, 0, 0, 0}, g3 = {0, 0, 0, 0};
#if __has_include(<hip/amd_detail/amd_gfx1250_TDM.h>)
    iv8 g4 = {0, 0, 0, 0, 0, 0, 0, 0};
    __builtin_amdgcn_tensor_store_from_lds(g0, g1, g2, g3, g4, 0);
#else
    __builtin_amdgcn_tensor_store_from_lds(g0, g1, g2, g3, 0);
#endif
}
#endif

__global__ __launch_bounds__(BLOCK, 1)
void qsoftmax_kernel(const int* __restrict__ data, const float* __restrict__ dscale,
                     float* __restrict__ out) {
    __shared__ int   tile[CCH][TILE];    // 128 KB slab: tile[c][pixel], reused for output
    __shared__ float expLut[256];        // exp_q as a function of diff = qmax - q

    const int t  = threadIdx.x;
    const int b  = blockIdx.y;
    const int p0 = blockIdx.x * TILE;
    const float ds = dscale[0];

    const size_t planeBase = (size_t)b * CCH * HWPX + (size_t)p0;
    const u32 ldsBase = (u32)(uintptr_t)&tile[0][0];   // generic addr low 32b == LDS offset

#if __has_builtin(__builtin_amdgcn_tensor_load_to_lds)
    // Wave 0 kicks off the DMA first so it overlaps the LUT build below.
    if (t < 32) tdm_load((uint64_t)(uintptr_t)(data + planeBase), ldsBase);
#endif

    // Build the exp LUT (expf + IEEE-exact quantizing divide) while DMA flies.
    expLut[t] = exp_pw_q(-(float)t * ds);

#if __has_builtin(__builtin_amdgcn_tensor_load_to_lds)
    if (t < 32) __builtin_amdgcn_s_wait_tensorcnt(0);
#else
    for (int i = t; i < CCH * TILE; i += BLOCK) {      // fallback cooperative load
        int c = i / TILE, p = i - c * TILE;
        (&tile[0][0])[i] = data[planeBase + (size_t)c * HWPX + p];
    }
#endif
    __syncthreads();

    // Pass 1: channel max for this thread's pixel (conflict-free LDS column).
    int qmax = tile[0][t];
#pragma unroll 8
    for (int c = 1; c < CCH; ++c) { int vI = tile[c][t]; qmax = vI > qmax ? vI : qmax; }

    // Pass 2: exact integer-valued sum of quantized exp over channels.
    float sum = 0.0f;
#pragma unroll 8
    for (int c = 0; c < CCH; ++c) {
        int d = qmax - tile[c][t];
        d = d > 255 ? 255 : d;
        sum += expLut[d];                 // <= 128*32767 < 2^24, exact in f32
    }

    // shift = ceil(log2(128)) = 7
    float sum_sh = fminf(fmaxf(rintf(sum * 0.0078125f), -32768.0f), 32767.0f);
    const float div_scale = (float)((2.0 / 65535.0) * 128.0);
    float rq = recip_pw_q(sum_sh * div_scale);

    // Pass 3: requantize exp*recip to int8 scale, write dequantized f32
    // IN PLACE into this thread's LDS column (each thread owns column t).
    const float K    = (float)((2.0 / 65535.0) * (20.0 / 65535.0) / (2.0 / 255.0));
    const float OUTS = (float)(2.0 / 255.0);
#pragma unroll 8
    for (int c = 0; c < CCH; ++c) {
        int d = qmax - tile[c][t];
        d = d > 255 ? 255 : d;
        float oq = fminf(fmaxf(rintf(expLut[d] * rq * K), -128.0f), 127.0f);
        tile[c][t] = __float_as_int(oq * OUTS);
    }
    __syncthreads();

#if __has_builtin(__builtin_amdgcn_tensor_load_to_lds)
    // Wave 0 DMA-stores the finished f32 slab back with the mirror descriptor.
    if (t < 32) {
        tdm_store((uint64_t)(uintptr_t)(out + planeBase), ldsBase);
        __builtin_amdgcn_s_wait_tensorcnt(0);
    }
#else
    for (int i = t; i < CCH * TILE; i += BLOCK) {      // fallback cooperative store
        int c = i / TILE, p = i - c * TILE;
        out[planeBase + (size_t)c * HWPX + p] = __int_as_float((&tile[0][0])[i]);
    }
#endif
}

extern "C" void kernel_launch(void* const* d_in, const int* in_sizes, int n_in,
                              void* d_out, int out_size, void* d_ws, size_t ws_size,
                              hipStream_t stream) {
    const int*   data = (const int*)d_in[0];
    const float* dsc  = (const float*)d_in[1];
    float*       out  = (float*)d_out;
    const int B = in_sizes[0] / (CCH * HWPX);     // 32 for the reference shapes
    dim3 grid(HWPX / TILE, B, 1);
    qsoftmax_kernel<<<grid, dim3(BLOCK, 1, 1), 0, stream>>>(data, dsc, out);
}